// NCEGroupCriterion_29128468201562
// MI455X (gfx1250) — compile-verified
//
#include <hip/hip_runtime.h>

// NCEGroupCriterion forward on MI455X (gfx1250).
//
// Memory-bound streaming reduction: 134 MB read once -> ~5.8 us at 23.3 TB/s.
// Per element: one log(x+c); we amortize 4 elements per v_log_f32 via
// log(prod of 4 factors). Wave-level reduction uses V_WMMA_F32_16X16X4_F32
// with an all-ones B matrix (exact f32 FMA reduction across lanes).

typedef __attribute__((ext_vector_type(2))) float v2f;
typedef __attribute__((ext_vector_type(8))) float v8f;

#define N_ROWS      8192
#define ROW_STRIDE  4097
#define K_COLS      4096          // last column (k==4096) never contributes
#define BLK         256

// Constants computed in double (matching Python host arithmetic), cast to f32.
#define C_ADD   ((float)(100.0 / 1281167.0 + 1e-7))
#define KPN     ((float)(100.0 / 1281167.0))
#define MINPROB 0.003f

// Exact f32 sum of all 32 lane values via WMMA:
//   A[m][0]=acc_m (lanes 0-15 vgpr0), A[m][2]=acc_{m+16} (lanes 16-31 vgpr0),
//   A[m][1]=A[m][3]=0, B=ones  =>  D[m][n] = acc_m + acc_{m+16}.
// Per-lane sum of the 8 D vgprs gives sum over M=0..7 (lanes<16) or M=8..15
// (lanes>=16); one shfl_xor(16) completes the wave sum in every lane.
// Requires EXEC all ones (call only from fully reconverged code).
__device__ __forceinline__ float wave_sum_wmma(float acc) {
    v2f a; a[0] = acc;  a[1] = 0.0f;
    v2f b; b[0] = 1.0f; b[1] = 1.0f;
    v8f c = {};
    c = __builtin_amdgcn_wmma_f32_16x16x4_f32(
            /*neg_a=*/false, a, /*neg_b=*/false, b,
            /*c_mod=*/(short)0, c, /*reuse_a=*/false, /*reuse_b=*/false);
    float t = ((c[0] + c[1]) + (c[2] + c[3])) + ((c[4] + c[5]) + (c[6] + c[7]));
    t += __shfl_xor(t, 16, 32);
    return t;
}

__global__ void __launch_bounds__(BLK)
nce_row_kernel(const float* __restrict__ x, float* __restrict__ partial) {
    const int  row  = blockIdx.x;
    const int  tid  = threadIdx.x;
    const long base = (long)row * ROW_STRIDE;

    __shared__ int   sj;
    __shared__ float swave[BLK / 32];

    if (tid == 0) {
        float x1 = x[base + 1], x2 = x[base + 2], x3 = x[base + 3];
        sj = (x1 < MINPROB) ? 1 : (x2 < MINPROB) ? 2 : (x3 < MINPROB) ? 3 : 4;
    }
    __syncthreads();
    const int j = sj;   // j in {1,2,3,4}

    float acc = 0.0f;

    // Head, k in [0, j): weight * (log(x) - log(x+c)).
    // Row 0 uses weight 1 (reference rebinds lnPmt each step); rows>0 use (j-k).
    if (tid < j) {
        float xv = x[base + tid];
        float w  = (row == 0) ? 1.0f : (float)(j - tid);
        acc += w * (__logf(xv) - __logf(xv + C_ADD));
    }
    // Columns [j, 4): part of the term_o tail handled scalar (k == tid).
    if (tid >= j && tid < 4) {
        acc -= __logf(x[base + tid] + C_ADD);
    }
    // Closed-form piece of term_o: (K - j) * log(K1*Pn), once per row.
    if (tid == 0) {
        acc += (float)(K_COLS - j) * __logf(KPN);
    }

    // Bulk tail, columns [4, 4096): coalesced strided b32 loads
    // (wave32 lanes read 128 contiguous bytes per vmem op).
    // Fold 4 strided factors into one v_log_f32: log(prod (x+c)).
    // Factors in [c, 1+c] -> prod of 4 in [3.7e-17, ~1]: no under/overflow.
    #pragma unroll
    for (int g = 0; g < 4; ++g) {
        float p = 1.0f;
        #pragma unroll
        for (int s = 0; s < 4; ++s) {
            int col = 4 + (g * 4 + s) * BLK + tid;          // up to 4099
            float v = 1.0f;                                  // log(1) == 0
            if (col < K_COLS) v = x[base + col] + C_ADD;
            p *= v;
        }
        acc -= __logf(p);
    }

    // Fully reconverged here: EXEC all ones -> WMMA wave reduction is legal.
    float wsum = wave_sum_wmma(acc);

    const int lane = tid & 31;
    const int wid  = tid >> 5;
    if (lane == 0) swave[wid] = wsum;
    __syncthreads();
    if (tid == 0) {
        float r = 0.0f;
        #pragma unroll
        for (int w = 0; w < BLK / 32; ++w) r += swave[w];
        partial[row] = r;
    }
}

// Deterministic final reduction of 8192 row partials in f64, then scale.
__global__ void __launch_bounds__(BLK)
nce_final_kernel(const float* __restrict__ partial, float* __restrict__ out) {
    __shared__ double sd[BLK];
    const int tid = threadIdx.x;

    double s = 0.0;
    for (int i = tid; i < N_ROWS; i += BLK) s += (double)partial[i];
    sd[tid] = s;
    __syncthreads();

    for (int off = BLK / 2; off > 0; off >>= 1) {
        if (tid < off) sd[tid] += sd[tid + off];
        __syncthreads();
    }
    if (tid == 0) out[0] = (float)(-sd[0] / (double)N_ROWS);
}

extern "C" void kernel_launch(void* const* d_in, const int* in_sizes, int n_in,
                              void* d_out, int out_size, void* d_ws, size_t ws_size,
                              hipStream_t stream) {
    const float* x       = (const float*)d_in[0];   // f32[8192, 4097]
    // d_in[1] (targets) is unused by the reference computation.
    float*       out     = (float*)d_out;           // scalar f32
    float*       partial = (float*)d_ws;            // 8192 f32 = 32 KB scratch

    nce_row_kernel<<<N_ROWS, BLK, 0, stream>>>(x, partial);
    nce_final_kernel<<<1, BLK, 0, stream>>>(partial, out);
}